// GCN_86646670230122
// MI455X (gfx1250) — compile-verified
//
#include <hip/hip_runtime.h>
#include <hip/hip_bf16.h>

// MI455X / gfx1250, wave32. fp32 GEMMs via V_WMMA_F32_16X16X4_F32.

typedef float v2f __attribute__((ext_vector_type(2)));
typedef float v8f __attribute__((ext_vector_type(8)));

#define BB 32
#define CC 128
#define LL 256

// ---------------------------------------------------------------------------
// K1: diff[b,i,j] = sum_l |x[b,i,l] - x[b,j,l]|
// grid: B * (C/16) * (C/16) = 2048 blocks, 256 threads.
// LDS rows padded to stride 260 floats: xj[jj][l] reads hit 16 distinct banks
// (jj*260 % 64 = jj*4, distinct for jj in 0..15); xi reads are broadcasts.
// ---------------------------------------------------------------------------
__global__ void k_diff(const float* __restrict__ x, float* __restrict__ diff) {
    const int blk = blockIdx.x;
    const int b  = blk >> 6;            // / 64
    const int it = (blk >> 3) & 7;      // i-tile
    const int jt = blk & 7;             // j-tile
    const int i0 = it * 16, j0 = jt * 16;

    __shared__ float xi[16 * 260];
    __shared__ float xj[16 * 260];

    const int tid = threadIdx.x;
    for (int f = tid; f < 16 * LL; f += 256) {
        const int r = f >> 8, l = f & 255;
        xi[r * 260 + l] = x[((b * CC + i0 + r) * LL) + l];
        xj[r * 260 + l] = x[((b * CC + j0 + r) * LL) + l];
    }
    __syncthreads();

    const int ii = tid >> 4, jj = tid & 15;
    float s = 0.0f;
    #pragma unroll 8
    for (int l = 0; l < LL; ++l)
        s += fabsf(xi[ii * 260 + l] - xj[jj * 260 + l]);

    diff[(b * CC + i0 + ii) * CC + j0 + jj] = s;
}

// ---------------------------------------------------------------------------
// K2: in-place: diff -> tmpS = exp(relu((1 - diff/||diff||_B) * a))
// grid: C*C/256 = 64 blocks. Coalesced over j within each batch slice.
// ---------------------------------------------------------------------------
__global__ void k_tmpS(float* __restrict__ buf, const float* __restrict__ a) {
    const int idx = blockIdx.x * 256 + threadIdx.x;   // over C*C
    float s = 0.0f;
    #pragma unroll
    for (int b = 0; b < BB; ++b) {
        const float d = buf[b * CC * CC + idx];
        s += d * d;
    }
    const float inv = 1.0f / fmaxf(sqrtf(s), 1e-12f);
    const float av  = a[idx];
    #pragma unroll
    for (int b = 0; b < BB; ++b) {
        const float d = buf[b * CC * CC + idx] * inv;
        buf[b * CC * CC + idx] = __expf(fmaxf((1.0f - d) * av, 0.0f));
    }
}

// ---------------------------------------------------------------------------
// K3: colsum[b,j] = sum_i tmpS[b,i,j]; deg[b,j] = sum_i tmpS[b,i,j]/colsum
// grid: B blocks of 128 threads (thread = j, coalesced row reads).
// ---------------------------------------------------------------------------
__global__ void k_colsum(const float* __restrict__ tmpS,
                         float* __restrict__ colsum, float* __restrict__ deg) {
    const int b = blockIdx.x, j = threadIdx.x;
    float s = 0.0f;
    for (int i = 0; i < CC; ++i) s += tmpS[(b * CC + i) * CC + j];
    colsum[b * CC + j] = s;
    const float r = 1.0f / s;
    float d = 0.0f;
    for (int i = 0; i < CC; ++i) d += tmpS[(b * CC + i) * CC + j] * r;
    deg[b * CC + j] = d;
}

// ---------------------------------------------------------------------------
// K4: LtT[b][j][i] = Lt[b][i][j] = (diag(deg) - adj - I)[i,j]   (transposed
// store so K5's WMMA A-fragments are contiguous float2 row loads).
// 16x16 LDS tile transpose (pad 17) keeps both read and write coalesced.
// grid: B * 8 * 8 blocks, 256 threads.
// ---------------------------------------------------------------------------
__global__ void k_lapT(const float* __restrict__ tmpS,
                       const float* __restrict__ colsum,
                       const float* __restrict__ deg,
                       float* __restrict__ ltT) {
    const int blk = blockIdx.x;
    const int b  = blk >> 6;
    const int it = (blk >> 3) & 7;
    const int jt = blk & 7;
    const int i0 = it * 16, j0 = jt * 16;

    __shared__ float t[16 * 17];
    const int tid = threadIdx.x;
    const int ii = tid >> 4, jj = tid & 15;

    const int gi = i0 + ii, gj = j0 + jj;
    float lt = -tmpS[(b * CC + gi) * CC + gj] / colsum[b * CC + gj];
    if (gi == gj) lt += deg[b * CC + gi] - 1.0f;
    t[jj * 17 + ii] = lt;                 // transpose in LDS
    __syncthreads();
    // row J = j0+ii of LtT, cols i0+jj : coalesced over jj
    ltT[(b * CC + j0 + ii) * CC + i0 + jj] = t[ii * 17 + jj];
}

// ---------------------------------------------------------------------------
// K5: rhs[b,j,l] = sum_i Lt[b,i,j] * x[b,i,l]   (Lt^T @ x per batch)
// grid: B * (L/16) = 512 blocks, 256 threads = 8 waves; wave w owns j-tile w.
// A-fragment: contiguous float2 from LtT row (one global_load_b64/step).
// B-operand x[b][:, l0..l0+15] staged pair-interleaved in LDS so each lane's
// B fragment (K=v row pair) is one 8-byte ds read, conflict-free.
// ---------------------------------------------------------------------------
__global__ void k_rhs(const float* __restrict__ ltT, const float* __restrict__ x,
                      float* __restrict__ rhs) {
    const int b  = blockIdx.x >> 4;
    const int l0 = (blockIdx.x & 15) * 16;

    __shared__ float xt[CC * 16];   // pair-interleaved: [(i>>1)*32 + n*2 + (i&1)]
    const int tid = threadIdx.x;
    for (int f = tid; f < CC * 16; f += 256) {
        const int i = f >> 4, n = f & 15;
        xt[(i >> 1) * 32 + n * 2 + (i & 1)] = x[(b * CC + i) * LL + l0 + n];
    }
    __syncthreads();

    const int wave = tid >> 5, lane = tid & 31;
    const int h = lane >> 4, ml = lane & 15;
    const int j0 = wave * 16;
    const float* arow = ltT + (b * CC + j0 + ml) * CC;  // row j0+ml of Lt^T

    v8f c = {};
    for (int t = 0; t < 32; ++t) {          // K = 128 = 32 * 4
        const int k = 4 * t + 2 * h;
        const v2f av = *(const v2f*)&arow[k];                 // A[m=ml][k,k+1]
        const v2f bv = *(const v2f*)&xt[(2 * t + h) * 32 + ml * 2];
        c = __builtin_amdgcn_wmma_f32_16x16x4_f32(false, av, false, bv,
                                                  (short)0, c, false, false);
    }
    float* o = rhs + (b * CC + j0 + 8 * h) * LL + l0 + ml;
    #pragma unroll
    for (int r = 0; r < 8; ++r) o[r * LL] = c[r];
}

// ---------------------------------------------------------------------------
// K6: out[b] = relu( x[b] @ Theta0 + rhs[b] @ Theta1 )   M=128,N=256,K=256
// grid: B * (256/16) = 512 blocks, 256 threads = 8 waves; wave w owns m-tile w.
// Theta column tiles (both k) staged pair-interleaved in 32 KB LDS.
// Single f32 accumulator chain: 128 WMMAs per 16x16 output tile.
// ---------------------------------------------------------------------------
__global__ void k_out(const float* __restrict__ x, const float* __restrict__ rhs,
                      const float* __restrict__ Theta, float* __restrict__ out) {
    const int b  = blockIdx.x >> 4;
    const int n0 = (blockIdx.x & 15) * 16;

    __shared__ float t0[LL * 16];
    __shared__ float t1[LL * 16];
    const int tid = threadIdx.x;
    for (int f = tid; f < LL * 16; f += 256) {
        const int l = f >> 4, n = f & 15;
        const int s = (l >> 1) * 32 + n * 2 + (l & 1);
        t0[s] = Theta[l * 256 + n0 + n];                 // Theta[0]
        t1[s] = Theta[LL * 256 + l * 256 + n0 + n];      // Theta[1]
    }
    __syncthreads();

    const int wave = tid >> 5, lane = tid & 31;
    const int h = lane >> 4, ml = lane & 15;
    const int m0 = wave * 16;
    const float* xrow = x   + (b * CC + m0 + ml) * LL;
    const float* rrow = rhs + (b * CC + m0 + ml) * LL;

    v8f c = {};
    for (int t = 0; t < 64; ++t) {          // K = 256 = 64 * 4
        const int k = 4 * t + 2 * h;
        const int p = (2 * t + h) * 32 + ml * 2;
        const v2f a0 = *(const v2f*)&xrow[k];           // contiguous float2 of row
        const v2f b0 = *(const v2f*)&t0[p];
        c = __builtin_amdgcn_wmma_f32_16x16x4_f32(false, a0, false, b0,
                                                  (short)0, c, false, false);
        const v2f a1 = *(const v2f*)&rrow[k];
        const v2f b1 = *(const v2f*)&t1[p];
        c = __builtin_amdgcn_wmma_f32_16x16x4_f32(false, a1, false, b1,
                                                  (short)0, c, false, false);
    }
    float* o = out + (b * CC + m0 + 8 * h) * 256 + n0 + ml;
    #pragma unroll
    for (int r = 0; r < 8; ++r) o[r * 256] = fmaxf(c[r], 0.0f);
}

// ---------------------------------------------------------------------------
extern "C" void kernel_launch(void* const* d_in, const int* in_sizes, int n_in,
                              void* d_out, int out_size, void* d_ws, size_t ws_size,
                              hipStream_t stream) {
    const float* x     = (const float*)d_in[0];   // [32,128,256]
    const float* a     = (const float*)d_in[1];   // [128,128]
    const float* Theta = (const float*)d_in[2];   // [2,256,256]
    float* out = (float*)d_out;                   // [32,128,256]

    // workspace: diff/tmpS (2MB, in place) | LtT (2MB) | colsum | deg | rhs (4MB)
    float* buf    = (float*)d_ws;                 // B*C*C
    float* ltT    = buf + BB * CC * CC;           // B*C*C (transposed Laplacian)
    float* colsum = ltT + BB * CC * CC;           // B*C
    float* deg    = colsum + BB * CC;             // B*C
    float* rhs    = deg + BB * CC;                // B*C*L

    k_diff  <<<BB * 8 * 8,            256, 0, stream>>>(x, buf);
    k_tmpS  <<<(CC * CC) / 256,       256, 0, stream>>>(buf, a);
    k_colsum<<<BB,                    CC,  0, stream>>>(buf, colsum, deg);
    k_lapT  <<<BB * 8 * 8,            256, 0, stream>>>(buf, colsum, deg, ltT);
    k_rhs   <<<BB * (LL / 16),        256, 0, stream>>>(ltT, x, rhs);
    k_out   <<<BB * (256 / 16),       256, 0, stream>>>(x, rhs, Theta, out);
}